// conv3d_65807488909370
// MI455X (gfx1250) — compile-verified
//
#include <hip/hip_runtime.h>

typedef __attribute__((ext_vector_type(2))) float v2f;
typedef __attribute__((ext_vector_type(8))) float v8f;

#define CIN  64
#define COUT 64
#define LDSP 68          // LDS row pitch (floats): 64 + 4 pad -> conflict-free b64 reads
#define NOFF 26
#define TILE_M 32        // pairs/rows per block

__device__ __forceinline__ void atomic_add_f32(float* p, float v) {
    __hip_atomic_fetch_add(p, v, __ATOMIC_RELAXED, __HIP_MEMORY_SCOPE_AGENT);
}

__device__ __forceinline__ v8f wmma_f32(v2f a, v2f b, v8f c) {
    return __builtin_amdgcn_wmma_f32_16x16x4_f32(false, a, false, b, (short)0, c,
                                                 false, false);
}

// ---------------------------------------------------------------------------
// Offset contributions: gather rows via imap, GEMM with W[k], scatter-add via omap.
// grid = (ceil(P/32), 26); block = 128 thr = 4 waves; wave w owns cols [16w,16w+16).
// ---------------------------------------------------------------------------
__global__ __launch_bounds__(128) void scatter_gemm(const float* __restrict__ feats,
                                                    const float* __restrict__ Wall, // [27][64][64]
                                                    const int* __restrict__ imap,   // [26][P]
                                                    const int* __restrict__ omap,   // [26][P]
                                                    float* __restrict__ out,
                                                    int P) {
    __shared__ float ldsA[TILE_M * LDSP];

    const int k    = blockIdx.y;                 // 0..25 (non-center offsets)
    const int kidx = (k < 13) ? k : k + 1;       // skip center slot 13
    const float* W = Wall + (size_t)kidx * CIN * COUT;
    const size_t kP = (size_t)k * P;

    const int lane = threadIdx.x & 31;
    const int wave = threadIdx.x >> 5;
    const int m    = lane & 15;
    const int half = lane >> 4;                  // K-pair selector
    const int koff = half << 1;
    const int n    = wave * 16 + m;              // output column this lane owns

    // Preload this wave's 64x16 weight slab as 16 v2f fragments.
    v2f bfrag[16];
#pragma unroll
    for (int j = 0; j < 16; ++j) {
        const int kk = 4 * j + koff;
        bfrag[j].x = W[kk * COUT + n];
        bfrag[j].y = W[(kk + 1) * COUT + n];
    }

    const int tileP0 = blockIdx.x * TILE_M;

    // Gather-stage A tiles (32 rows x 64 floats): thread t -> row t/4, 16 floats.
    {
        const int ar = threadIdx.x >> 2;         // 0..31
        const int ac = (threadIdx.x & 3) * 16;   // 0,16,32,48
        const int p  = tileP0 + ar;
        int srcRow = -1;
        if (p < P) srcRow = imap[kP + p];
        float4 v0 = make_float4(0.f, 0.f, 0.f, 0.f), v1 = v0, v2 = v0, v3 = v0;
        if (srcRow >= 0) {
            const float4* src = (const float4*)(feats + (size_t)srcRow * CIN + ac);
            v0 = src[0]; v1 = src[1]; v2 = src[2]; v3 = src[3];
        }
        float* dst = &ldsA[ar * LDSP + ac];
        *(float4*)(dst)      = v0;
        *(float4*)(dst + 4)  = v1;
        *(float4*)(dst + 8)  = v2;
        *(float4*)(dst + 12) = v3;
    }
    __syncthreads();

    // Two 16x16 output tiles per wave: rows [0,16) and [16,32) of the block tile.
    v8f acc0 = {}, acc1 = {};
#pragma unroll
    for (int j = 0; j < 16; ++j) {
        const int kk = 4 * j + koff;
        v2f a0 = *(const v2f*)&ldsA[m * LDSP + kk];
        v2f a1 = *(const v2f*)&ldsA[(m + 16) * LDSP + kk];
        acc0 = wmma_f32(a0, bfrag[j], acc0);
        acc1 = wmma_f32(a1, bfrag[j], acc1);
    }

    // Scatter-add. D layout: VGPR j -> pair p = tileP0 + j + 8*half (+16 for acc1).
    const int pBase = tileP0 + (half << 3);
    if (tileP0 + TILE_M <= P) {
#pragma unroll
        for (int j = 0; j < 8; ++j) {
            const int d0 = omap[kP + pBase + j];
            if (d0 >= 0) atomic_add_f32(&out[(size_t)d0 * COUT + n], acc0[j]);
            const int d1 = omap[kP + pBase + j + 16];
            if (d1 >= 0) atomic_add_f32(&out[(size_t)d1 * COUT + n], acc1[j]);
        }
    } else {
#pragma unroll
        for (int j = 0; j < 8; ++j) {
            const int p0 = pBase + j;
            if (p0 < P) {
                const int d0 = omap[kP + p0];
                if (d0 >= 0) atomic_add_f32(&out[(size_t)d0 * COUT + n], acc0[j]);
            }
            const int p1 = pBase + j + 16;
            if (p1 < P) {
                const int d1 = omap[kP + p1];
                if (d1 >= 0) atomic_add_f32(&out[(size_t)d1 * COUT + n], acc1[j]);
            }
        }
    }
}

// ---------------------------------------------------------------------------
// Center contribution: out[r, :] = feats[r, :] @ W13   (plain stores, no atomics)
// ---------------------------------------------------------------------------
__global__ __launch_bounds__(128) void center_gemm(const float* __restrict__ feats,
                                                   const float* __restrict__ W, // [64][64]
                                                   float* __restrict__ out,
                                                   int Nrows) {
    __shared__ float ldsA[TILE_M * LDSP];

    const int lane = threadIdx.x & 31;
    const int wave = threadIdx.x >> 5;
    const int m    = lane & 15;
    const int half = lane >> 4;
    const int koff = half << 1;
    const int n    = wave * 16 + m;

    v2f bfrag[16];
#pragma unroll
    for (int j = 0; j < 16; ++j) {
        const int kk = 4 * j + koff;
        bfrag[j].x = W[kk * COUT + n];
        bfrag[j].y = W[(kk + 1) * COUT + n];
    }

    const int tileRow0 = blockIdx.x * TILE_M;

    {
        const int ar   = threadIdx.x >> 2;
        const int ac   = (threadIdx.x & 3) * 16;
        const int grow = tileRow0 + ar;
        float4 v0 = make_float4(0.f, 0.f, 0.f, 0.f), v1 = v0, v2 = v0, v3 = v0;
        if (grow < Nrows) {
            const float4* src = (const float4*)(feats + (size_t)grow * CIN + ac);
            v0 = src[0]; v1 = src[1]; v2 = src[2]; v3 = src[3];
        }
        float* dst = &ldsA[ar * LDSP + ac];
        *(float4*)(dst)      = v0;
        *(float4*)(dst + 4)  = v1;
        *(float4*)(dst + 8)  = v2;
        *(float4*)(dst + 12) = v3;
    }
    __syncthreads();

    v8f acc0 = {}, acc1 = {};
#pragma unroll
    for (int j = 0; j < 16; ++j) {
        const int kk = 4 * j + koff;
        v2f a0 = *(const v2f*)&ldsA[m * LDSP + kk];
        v2f a1 = *(const v2f*)&ldsA[(m + 16) * LDSP + kk];
        acc0 = wmma_f32(a0, bfrag[j], acc0);
        acc1 = wmma_f32(a1, bfrag[j], acc1);
    }

    // D layout: VGPR j -> row tileRow0 + j + 8*half (+16 for acc1), col n.
    const int row0 = tileRow0 + (half << 3);
    float* outp = out + (size_t)row0 * COUT + n;
    if (tileRow0 + TILE_M <= Nrows) {
#pragma unroll
        for (int j = 0; j < 8; ++j) {
            outp[j * COUT]        = acc0[j];
            outp[(j + 16) * COUT] = acc1[j];
        }
    } else {
#pragma unroll
        for (int j = 0; j < 8; ++j) {
            if (row0 + j < Nrows)      outp[j * COUT]        = acc0[j];
            if (row0 + j + 16 < Nrows) outp[(j + 16) * COUT] = acc1[j];
        }
    }
}

// ---------------------------------------------------------------------------
extern "C" void kernel_launch(void* const* d_in, const int* in_sizes, int n_in,
                              void* d_out, int out_size, void* d_ws, size_t ws_size,
                              hipStream_t stream) {
    const float* feats = (const float*)d_in[0];   // [N,64] fp32
    const float* kern  = (const float*)d_in[1];   // [27,64,64] fp32
    const int*   imap  = (const int*)d_in[2];     // [26,P] int32
    const int*   omap  = (const int*)d_in[3];     // [26,P] int32
    float*       out   = (float*)d_out;           // [N,64] fp32

    const int N = in_sizes[0] / CIN;
    const int P = in_sizes[2] / NOFF;

    const int tilesN = (N + TILE_M - 1) / TILE_M;
    center_gemm<<<dim3(tilesN), dim3(128), 0, stream>>>(
        feats, kern + (size_t)13 * CIN * COUT, out, N);

    const int tilesP = (P + TILE_M - 1) / TILE_M;
    scatter_gemm<<<dim3(tilesP, NOFF), dim3(128), 0, stream>>>(
        feats, kern, imap, omap, out, P);
}